// GAT_60413009985601
// MI455X (gfx1250) — compile-verified
//
#include <hip/hip_runtime.h>
#include <math.h>

// ---------------------------------------------------------------------------
// GAT (2-layer, single head) for MI455X / gfx1250.
//   N=100000 nodes (multiple of 16), E=3200000 edges (+N self-loops).
//   Layer 1: 2 -> 16, Layer 2: 16 -> 64.
// Dense linears use V_WMMA_F32_16X16X4_F32 (exact f32 matrix path).
// Sparse softmax/aggregation: L2-resident gather + float atomics.
// ---------------------------------------------------------------------------

#define N_NODES   100000
#define E_EDGES   3200000
#define ETOT      (E_EDGES + N_NODES)   // 3,300,000
#define NEG_SLOPE 0.2f

typedef __attribute__((ext_vector_type(2))) float v2f;
typedef __attribute__((ext_vector_type(8))) float v8f;

static __device__ __forceinline__ float lrelu(float v) {
    return v > 0.0f ? v : v * NEG_SLOPE;
}

// float atomic max via sign-split int/uint atomics (init value must be -inf)
static __device__ __forceinline__ void atomicMaxF(float* addr, float v) {
    if (v >= 0.0f)
        atomicMax((int*)addr, __float_as_int(v));
    else
        atomicMin((unsigned int*)addr, __float_as_uint(v));
}

// ---------------------------------------------------------------------------
// H[n, FOUT] = X[n, FIN] @ W[FIN, FOUT]   via V_WMMA_F32_16X16X4_F32
// One wave handles a 16-row tile of nodes; K padded up to multiple of 4.
// Requires n % 16 == 0 (true: 100000 = 6250*16) -> fully branch-free body.
// A layout (16x4 f32): lanes 0-15 give K=0 (v0) / K=1 (v1) for row M=lane;
//                      lanes 16-31 give K=2 / K=3 for row M=lane-16.
// B layout (4x16 f32): v0 = row K=0 (lanes 0-15) / K=2 (lanes 16-31),
//                      v1 = row K=1 / K=3, column = lane%16.
// D layout (16x16 f32): VGPR i -> row m0+i (lanes 0-15) / m0+i+8 (lanes 16-31).
// K-padding handled with clamped (always-valid) addresses + value selects,
// so no exec-mask branching is generated.
// ---------------------------------------------------------------------------
template <int FIN, int FOUT>
__global__ __launch_bounds__(256) void linear_wmma(const float* __restrict__ X,
                                                   const float* __restrict__ W,
                                                   float* __restrict__ H, int n) {
    const int lane = threadIdx.x & 31;
    const int wave = threadIdx.x >> 5;
    const int tile = blockIdx.x * (blockDim.x >> 5) + wave;
    const int m0   = tile * 16;
    if (m0 >= n) return;                       // wave-uniform exit

    const int half = lane >> 4;                // 0 or 1
    const int l16  = lane & 15;
    const int row  = m0 + l16;

    constexpr int KSTEPS = (FIN + 3) / 4;
    constexpr int NTILES = FOUT / 16;

    const v2f* __restrict__ X2 = reinterpret_cast<const v2f*>(X);

    v2f afrag[KSTEPS];
#pragma unroll
    for (int ks = 0; ks < KSTEPS; ++ks) {
        const int k0  = ks * 4 + half * 2;     // even; pair (k0, k0+1) contiguous
        const int kc  = (k0 < FIN) ? k0 : 0;   // clamp -> address always valid
        v2f a = X2[(row * FIN + kc) >> 1];     // single b64 load
        a[0] = (k0     < FIN) ? a[0] : 0.0f;   // value selects (v_cndmask)
        a[1] = (k0 + 1 < FIN) ? a[1] : 0.0f;
        afrag[ks] = a;
    }

#pragma unroll
    for (int j = 0; j < NTILES; ++j) {
        const int col = j * 16 + l16;
        v8f c = {0.f, 0.f, 0.f, 0.f, 0.f, 0.f, 0.f, 0.f};
#pragma unroll
        for (int ks = 0; ks < KSTEPS; ++ks) {
            const int k0  = ks * 4 + half * 2;
            const int kr0 = (k0     < FIN) ? k0     : 0;
            const int kr1 = (k0 + 1 < FIN) ? k0 + 1 : 0;
            v2f b;
            b[0] = W[kr0 * FOUT + col];
            b[1] = W[kr1 * FOUT + col];
            b[0] = (k0     < FIN) ? b[0] : 0.0f;
            b[1] = (k0 + 1 < FIN) ? b[1] : 0.0f;
            c = __builtin_amdgcn_wmma_f32_16x16x4_f32(
                    /*neg_a=*/false, afrag[ks], /*neg_b=*/false, b,
                    /*c_mod=*/(short)0, c, /*reuse_a=*/false, /*reuse_b=*/false);
        }
#pragma unroll
        for (int i = 0; i < 8; ++i) {
            H[(m0 + i + half * 8) * FOUT + col] = c[i];   // unconditional stores
        }
    }
}

// ---------------------------------------------------------------------------
// Per-node prep: alpha_src/alpha_dst dots, init m=-inf, s=0, zero accumulator.
// ---------------------------------------------------------------------------
template <int FOUT>
__global__ __launch_bounds__(256) void node_prep(const float* __restrict__ H,
                                                 const float* __restrict__ a_src,
                                                 const float* __restrict__ a_dst,
                                                 float* __restrict__ as_,
                                                 float* __restrict__ ad_,
                                                 float* __restrict__ m_,
                                                 float* __restrict__ s_,
                                                 float* __restrict__ agg, int n) {
    const int i = blockIdx.x * blockDim.x + threadIdx.x;
    if (i >= n) return;
    float ssum = 0.0f, dsum = 0.0f;
#pragma unroll
    for (int f = 0; f < FOUT; ++f) {
        const float h = H[i * FOUT + f];
        ssum += h * a_src[f];
        dsum += h * a_dst[f];
        agg[i * FOUT + f] = 0.0f;
    }
    as_[i] = ssum;
    ad_[i] = dsum;
    m_[i]  = -INFINITY;
    s_[i]  = 0.0f;
}

// ---------------------------------------------------------------------------
// Edge pass 1: segment max of leaky-relu logits into m[dst].
// Self-loops synthesized for e >= E_EDGES.
// ---------------------------------------------------------------------------
__global__ __launch_bounds__(256) void edge_max(const int* __restrict__ srcIdx,
                                                const int* __restrict__ dstIdx,
                                                const float* __restrict__ as_,
                                                const float* __restrict__ ad_,
                                                float* __restrict__ m_) {
    const int e = blockIdx.x * blockDim.x + threadIdx.x;
    if (e >= ETOT) return;
    int s, d;
    if (e < E_EDGES) { s = srcIdx[e]; d = dstIdx[e]; }
    else             { s = d = e - E_EDGES; }
    atomicMaxF(&m_[d], lrelu(as_[s] + ad_[d]));
}

// ---------------------------------------------------------------------------
// Edge pass 2: p = exp(e - m[dst]); s[dst] += p; store unnormalized p.
// ---------------------------------------------------------------------------
__global__ __launch_bounds__(256) void edge_exp(const int* __restrict__ srcIdx,
                                                const int* __restrict__ dstIdx,
                                                const float* __restrict__ as_,
                                                const float* __restrict__ ad_,
                                                const float* __restrict__ m_,
                                                float* __restrict__ s_,
                                                float* __restrict__ p_) {
    const int e = blockIdx.x * blockDim.x + threadIdx.x;
    if (e >= ETOT) return;
    int s, d;
    if (e < E_EDGES) { s = srcIdx[e]; d = dstIdx[e]; }
    else             { s = d = e - E_EDGES; }
    const float v  = lrelu(as_[s] + ad_[d]);
    const float pe = expf(v - m_[d]);
    p_[e] = pe;
    atomicAdd(&s_[d], pe);
}

// ---------------------------------------------------------------------------
// Edge pass 3: normalize p in place: alpha = p / s[dst].
// ---------------------------------------------------------------------------
__global__ __launch_bounds__(256) void edge_norm(const int* __restrict__ dstIdx,
                                                 const float* __restrict__ s_,
                                                 float* __restrict__ p_) {
    const int e = blockIdx.x * blockDim.x + threadIdx.x;
    if (e >= ETOT) return;
    const int d = (e < E_EDGES) ? dstIdx[e] : (e - E_EDGES);
    p_[e] = p_[e] / s_[d];
}

// ---------------------------------------------------------------------------
// Edge pass 4: agg[dst, f] += alpha[e] * H[src, f]; one thread per (edge, f).
// Gathers and atomics stay resident in the 192MB L2 (node arrays < 60MB).
// ---------------------------------------------------------------------------
template <int FOUT>
__global__ __launch_bounds__(256) void edge_scatter(const int* __restrict__ srcIdx,
                                                    const int* __restrict__ dstIdx,
                                                    const float* __restrict__ p_,
                                                    const float* __restrict__ H,
                                                    float* __restrict__ agg) {
    const long long tid   = (long long)blockIdx.x * blockDim.x + threadIdx.x;
    const long long total = (long long)ETOT * FOUT;
    if (tid >= total) return;
    const int f = (int)(tid & (FOUT - 1));
    const int e = (int)(tid / FOUT);
    int s, d;
    if (e < E_EDGES) { s = srcIdx[e]; d = dstIdx[e]; }
    else             { s = d = e - E_EDGES; }
    atomicAdd(&agg[d * FOUT + f], p_[e] * H[s * FOUT + f]);
}

// ---------------------------------------------------------------------------
// out = relu(agg + bias)  (in place)
// ---------------------------------------------------------------------------
template <int FOUT>
__global__ __launch_bounds__(256) void bias_relu(float* __restrict__ agg,
                                                 const float* __restrict__ b, int n) {
    const int t = blockIdx.x * blockDim.x + threadIdx.x;
    if (t >= n * FOUT) return;
    const float v = agg[t] + b[t & (FOUT - 1)];
    agg[t] = v > 0.0f ? v : 0.0f;
}

// ---------------------------------------------------------------------------
extern "C" void kernel_launch(void* const* d_in, const int* in_sizes, int n_in,
                              void* d_out, int out_size, void* d_ws, size_t ws_size,
                              hipStream_t stream) {
    (void)in_sizes; (void)n_in; (void)out_size; (void)ws_size;

    const float* x      = (const float*)d_in[0];           // [N,2]
    const int*   ei     = (const int*)d_in[1];             // [2,E]
    const float* W1     = (const float*)d_in[2];           // [2,16]
    const float* a_src1 = (const float*)d_in[3];
    const float* a_dst1 = (const float*)d_in[4];
    const float* b1     = (const float*)d_in[5];
    const float* W2     = (const float*)d_in[6];           // [16,64]
    const float* a_src2 = (const float*)d_in[7];
    const float* a_dst2 = (const float*)d_in[8];
    const float* b2     = (const float*)d_in[9];

    const int* srcIdx = ei;             // edge_index[0]
    const int* dstIdx = ei + E_EDGES;   // edge_index[1]

    // workspace layout (floats)
    float* ws    = (float*)d_ws;
    float* hlin1 = ws;                            // N*16
    float* agg1  = hlin1 + (size_t)N_NODES * 16;  // N*16
    float* hlin2 = agg1  + (size_t)N_NODES * 16;  // N*64
    float* as_   = hlin2 + (size_t)N_NODES * 64;  // N
    float* ad_   = as_ + N_NODES;                 // N
    float* m_    = ad_ + N_NODES;                 // N
    float* s_    = m_  + N_NODES;                 // N
    float* p_    = s_  + N_NODES;                 // ETOT
    float* agg2  = (float*)d_out;                 // N*64

    const int TPB = 256;
    const int nodeBlocks = (N_NODES + TPB - 1) / TPB;
    const int edgeBlocks = (ETOT + TPB - 1) / TPB;
    const int tiles      = (N_NODES + 15) / 16;
    const int gemmBlocks = (tiles + 7) / 8;       // 8 waves per block

    // ---------------- Layer 1: 2 -> 16 ----------------
    linear_wmma<2, 16><<<gemmBlocks, TPB, 0, stream>>>(x, W1, hlin1, N_NODES);
    node_prep<16><<<nodeBlocks, TPB, 0, stream>>>(hlin1, a_src1, a_dst1,
                                                  as_, ad_, m_, s_, agg1, N_NODES);
    edge_max<<<edgeBlocks, TPB, 0, stream>>>(srcIdx, dstIdx, as_, ad_, m_);
    edge_exp<<<edgeBlocks, TPB, 0, stream>>>(srcIdx, dstIdx, as_, ad_, m_, s_, p_);
    edge_norm<<<edgeBlocks, TPB, 0, stream>>>(dstIdx, s_, p_);
    {
        const long long total = (long long)ETOT * 16;
        const int blocks = (int)((total + TPB - 1) / TPB);
        edge_scatter<16><<<blocks, TPB, 0, stream>>>(srcIdx, dstIdx, p_, hlin1, agg1);
    }
    bias_relu<16><<<(N_NODES * 16 + TPB - 1) / TPB, TPB, 0, stream>>>(agg1, b1, N_NODES);

    // ---------------- Layer 2: 16 -> 64 ----------------
    linear_wmma<16, 64><<<gemmBlocks, TPB, 0, stream>>>(agg1, W2, hlin2, N_NODES);
    node_prep<64><<<nodeBlocks, TPB, 0, stream>>>(hlin2, a_src2, a_dst2,
                                                  as_, ad_, m_, s_, agg2, N_NODES);
    edge_max<<<edgeBlocks, TPB, 0, stream>>>(srcIdx, dstIdx, as_, ad_, m_);
    edge_exp<<<edgeBlocks, TPB, 0, stream>>>(srcIdx, dstIdx, as_, ad_, m_, s_, p_);
    edge_norm<<<edgeBlocks, TPB, 0, stream>>>(dstIdx, s_, p_);
    {
        const long long total = (long long)ETOT * 64;
        const int blocks = (int)((total + TPB - 1) / TPB);
        edge_scatter<64><<<blocks, TPB, 0, stream>>>(srcIdx, dstIdx, p_, hlin2, agg2);
    }
    bias_relu<64><<<(N_NODES * 64 + TPB - 1) / TPB, TPB, 0, stream>>>(agg2, b2, N_NODES);
}